// LandmarksLoss_82145544503653
// MI455X (gfx1250) — compile-verified
//
#include <hip/hip_runtime.h>

#define DELTA   128
#define HALF    64
#define NB      16
#define NL      68
#define NH      224
#define NW      224
#define NSLICES (NB * NL)   // 1088
#define QW      (NW / 4)    // 56 float4 per row
#define NQ      (NH * QW)   // 12544 float4 per slice
#define BLOCK   256
#define TOTAL_N (54591488.0) // 16*68*224*224

typedef __attribute__((ext_vector_type(2))) float v2f;
typedef __attribute__((ext_vector_type(8))) float v8f;
typedef __attribute__((ext_vector_type(4))) int   v4i;

typedef __attribute__((address_space(1))) v4i* gptr_v4i;
typedef __attribute__((address_space(3))) v4i* lptr_v4i;

#if __has_builtin(__builtin_amdgcn_global_load_async_to_lds_b128)
#define HAS_ASYNC_LDS 1
#else
#define HAS_ASYNC_LDS 0
#endif

// Wave32 sum-reduction using V_WMMA_F32_16X16X4_F32 with B == ones:
// A[m][k] holds the 32 lane partials; D[m][n] = sum_k A[m][k]. Summing a
// lane's 8 D VGPRs gives sum over M=0..7 (lanes 0-15) or M=8..15 (lanes
// 16-31); one xor-16 shuffle completes the wave total.
__device__ __forceinline__ float wave_reduce_sum(float acc) {
#if __has_builtin(__builtin_amdgcn_wmma_f32_16x16x4_f32)
    v2f a;  a.x = acc;  a.y = 0.0f;
    v2f bo; bo.x = 1.0f; bo.y = 1.0f;
    v8f c = {};
    v8f d = __builtin_amdgcn_wmma_f32_16x16x4_f32(
        /*neg_a=*/false, a, /*neg_b=*/false, bo,
        /*c_mod=*/(short)0, c, /*reuse_a=*/false, /*reuse_b=*/false);
    float t = d[0] + d[1] + d[2] + d[3] + d[4] + d[5] + d[6] + d[7];
    t += __shfl_xor(t, 16, 32);
    return t;
#else
    for (int off = 16; off > 0; off >>= 1) acc += __shfl_xor(acc, off, 32);
    return acc;
#endif
}

// Branch-free masked bell fetch: clamp (v_med3), unconditional ds_load_b32,
// then v_cndmask to zero — mirrors the reference's clip+mask without EXEC churn.
__device__ __forceinline__ float bell_at(const float* __restrict__ brow,
                                         int by, bool rowok) {
    int byc = min(max(by, 0), DELTA - 1);
    float v = brow[byc];
    return (rowok && (unsigned)by < (unsigned)DELTA) ? v : 0.0f;
}

__global__ void __launch_bounds__(BLOCK)
landmarks_loss_main(const float* __restrict__ pred,
                    const float* __restrict__ lmk,
                    const float* __restrict__ bell,
                    float* __restrict__ partials) {
    __shared__ __align__(16) float bell_lds[DELTA * DELTA];  // 64 KB
    __shared__ float wsum[BLOCK / 32];

    // ---- Stage the 64 KB bell into LDS (async DMA path) ----
    {
        const float4* g = (const float4*)bell;
        float4*       l = (float4*)bell_lds;
        #pragma unroll
        for (int k = 0; k < (DELTA * DELTA) / 4 / BLOCK; ++k) {
            const int i = threadIdx.x + k * BLOCK;
#if HAS_ASYNC_LDS
            __builtin_amdgcn_global_load_async_to_lds_b128(
                (gptr_v4i)(g + i), (lptr_v4i)(l + i), /*offset=*/0, /*cpol=*/0);
#else
            l[i] = g[i];
#endif
        }
#if HAS_ASYNC_LDS
#if __has_builtin(__builtin_amdgcn_s_wait_asynccnt)
        __builtin_amdgcn_s_wait_asynccnt(0);
#else
        asm volatile("s_wait_asynccnt 0" ::: "memory");
#endif
#endif
    }
    __syncthreads();

    // ---- Stream this slice of pred, subtract shifted bell, accumulate ----
    const int slice = blockIdx.x;                       // b*L + l
    const float4* ps = (const float4*)(pred + (size_t)slice * (NH * NW));

    const int y_r = (int)rintf(lmk[slice * 2 + 0]);     // round-half-even, like jnp.round
    const int x_r = (int)rintf(lmk[slice * 2 + 1]);

    float acc = 0.0f;
    // i = threadIdx.x + k*BLOCK; maintain (h, wq) incrementally:
    // BLOCK = 4*QW + 32  ->  h += 4, wq += 32, carry once if wq >= QW.
    int h  = (int)threadIdx.x / QW;
    int wq = (int)threadIdx.x - h * QW;
    for (int i = threadIdx.x; i < NQ; i += BLOCK) {
        const float4 p = ps[i];                          // global_load_b128

        const int bx  = h - x_r + HALF;                  // uniform across a row
        const bool rowok = (unsigned)bx < (unsigned)DELTA;
        const int bxc = min(max(bx, 0), DELTA - 1);
        const float* brow = bell_lds + bxc * DELTA;
        const int by0 = wq * 4 - y_r + HALF;

        const float tx = bell_at(brow, by0 + 0, rowok);
        const float ty = bell_at(brow, by0 + 1, rowok);
        const float tz = bell_at(brow, by0 + 2, rowok);
        const float tw = bell_at(brow, by0 + 3, rowok);

        const float d0 = p.x - tx, d1 = p.y - ty, d2 = p.z - tz, d3 = p.w - tw;
        acc = fmaf(d0, d0, acc);
        acc = fmaf(d1, d1, acc);
        acc = fmaf(d2, d2, acc);
        acc = fmaf(d3, d3, acc);

        h  += 4;
        wq += 32;
        if (wq >= QW) { wq -= QW; h += 1; }
    }

    // ---- Block reduction: WMMA wave reduce, then LDS combine ----
    const float wtot = wave_reduce_sum(acc);            // EXEC all-1s here
    const int lane = threadIdx.x & 31;
    const int wv   = threadIdx.x >> 5;
    if (lane == 0) wsum[wv] = wtot;
    __syncthreads();
    if (threadIdx.x == 0) {
        float s = 0.0f;
        #pragma unroll
        for (int k = 0; k < BLOCK / 32; ++k) s += wsum[k];
        partials[slice] = s;
    }
}

__global__ void __launch_bounds__(BLOCK)
landmarks_loss_finalize(const float* __restrict__ partials,
                        float* __restrict__ out) {
    __shared__ float s[BLOCK];
    float a = 0.0f;
    for (int i = threadIdx.x; i < NSLICES; i += BLOCK) a += partials[i];
    s[threadIdx.x] = a;
    __syncthreads();
    for (int off = BLOCK / 2; off > 0; off >>= 1) {
        if (threadIdx.x < off) s[threadIdx.x] += s[threadIdx.x + off];
        __syncthreads();
    }
    if (threadIdx.x == 0) out[0] = s[0] * (float)(1.0 / TOTAL_N);
}

extern "C" void kernel_launch(void* const* d_in, const int* in_sizes, int n_in,
                              void* d_out, int out_size, void* d_ws, size_t ws_size,
                              hipStream_t stream) {
    (void)in_sizes; (void)n_in; (void)out_size; (void)ws_size;
    const float* pred = (const float*)d_in[0];   // (16,68,224,224) fp32
    const float* lmk  = (const float*)d_in[1];   // (16,68,2) fp32
    const float* bell = (const float*)d_in[2];   // (128,128) fp32
    float* out      = (float*)d_out;             // scalar fp32
    float* partials = (float*)d_ws;              // NSLICES floats, rewritten every call

    landmarks_loss_main<<<NSLICES, BLOCK, 0, stream>>>(pred, lmk, bell, partials);
    landmarks_loss_finalize<<<1, BLOCK, 0, stream>>>(partials, out);
}